// SAGC_pre_14233521619109
// MI455X (gfx1250) — compile-verified
//
#include <hip/hip_runtime.h>
#include <cstddef>

// Problem constants
#define Nn 64
#define Cc 64
#define Tt 256
#define Vv 25
#define Ss 3
#define Rr 8
#define Gg 8
#define CGc 8          // Cc/Gg
#define KK 9
#define TV_ 6400       // Tt*Vv
#define CTV_ 409600    // Cc*TV_

typedef float v2f __attribute__((ext_vector_type(2)));
typedef float v8f __attribute__((ext_vector_type(8)));
typedef unsigned int v4u __attribute__((ext_vector_type(4)));
typedef int v4i __attribute__((ext_vector_type(4)));
typedef int v8i __attribute__((ext_vector_type(8)));

// Workspace layout (float offsets)
#define OFF_XM  0                // N*C*V           = 102400
#define OFF_X1  102400           // S*N*R*V         = 38400
#define OFF_X2  140800           // S*N*R*V         = 38400
#define OFF_ADJ 179200           // S*N*C*V*V       = 7680000  (reused as z later)
#define OFF_Y   7859200          // N*C*T*V         = 6553600
#define OFF_YR  14412800         // N*C*T*V         = 6553600
#define OFF_ST  20966400         // stats           = 1280
// stats: [0,64) bn1 sum; [64,128) bn1 sq; [128,640) gn sum (n*8+g);
//        [640,1152) gn sq; [1152,1216) bn2 sum; [1216,1280) bn2 sq

// ---- Tensor Data Mover helpers (CDNA5 TDM, CDNA5 ISA) ----
__device__ __forceinline__ unsigned lds_off(const void* p) {
  // generic LDS address: addr[31:0] is the LDS byte offset (ISA 10.2 aperture table)
  return (unsigned)(unsigned long long)p;
}

// 1-D contiguous tile: len elements of 4 bytes, global -> LDS
__device__ __forceinline__ void tdm_load_1d(unsigned lds_addr, unsigned long long gaddr, int len) {
  // D# group0: [1:0]=count=1, [63:32]=lds_addr, [120:64]=global_addr, [127:126]=type=2
  v4u g0 = { 1u, lds_addr, (unsigned)gaddr,
             (unsigned)((gaddr >> 32) & 0x01FFFFFFull) | 0x80000000u };
  // D# group1: data_size=2(4B) @ [17:16]; tensor_dim0 @ [79:48]; tensor_dim1 @ [111:80];
  //            tile_dim0 @ [127:112]; tile_dim1 @ [143:128]; tensor_dim0_stride @ [207:160]
  v8i g1;
  g1[0] = 0x20000;                               // data_size = 4 bytes
  g1[1] = (len & 0xFFFF) << 16;                  // tensor_dim0 lo16
  g1[2] = ((len >> 16) & 0xFFFF) | (1 << 16);    // tensor_dim0 hi16 | tensor_dim1 = 1
  g1[3] = (len & 0xFFFF) << 16;                  // tensor_dim1 hi = 0 | tile_dim0 = len
  g1[4] = 1;                                     // tile_dim1 = 1, tile_dim2 = 0
  g1[5] = len;                                   // tensor_dim0_stride lo32
  g1[6] = 0;
  g1[7] = 0;
  v4i gz = { 0, 0, 0, 0 };
  v8i gz8 = { 0, 0, 0, 0, 0, 0, 0, 0 };
  __builtin_amdgcn_tensor_load_to_lds(g0, g1, gz, gz, gz8, 0);
}

__global__ __launch_bounds__(256) void k0_zero(float* __restrict__ st) {
  int i = blockIdx.x * 256 + threadIdx.x;
  if (i < 1280) st[i] = 0.f;
}

// xm[n,c,v] = mean_t x[n,c,t,v]
__global__ __launch_bounds__(256) void k1_mean(const float* __restrict__ x, float* __restrict__ ws) {
  int idx = blockIdx.x * 256 + threadIdx.x;       // N*C*V = 102400
  if (idx >= Nn * Cc * Vv) return;
  int v = idx % Vv;
  int nc = idx / Vv;                              // n*Cc + c
  const float* p = x + (size_t)nc * TV_ + v;
  float s = 0.f;
  for (int t = 0; t < Tt; ++t) s += p[t * Vv];
  ws[OFF_XM + idx] = s * (1.f / Tt);
}

// x1[i,n,r,v], x2[i,n,r,v] : 1x1 convs on xm
__global__ __launch_bounds__(256) void k2_x12(const float* __restrict__ w1, const float* __restrict__ b1,
                                              const float* __restrict__ w2, const float* __restrict__ b2,
                                              float* __restrict__ ws) {
  int idx = blockIdx.x * 256 + threadIdx.x;       // 2*S*N*R*V = 76800
  if (idx >= 2 * Ss * Nn * Rr * Vv) return;
  int which = idx / (Ss * Nn * Rr * Vv);
  int r0 = idx % (Ss * Nn * Rr * Vv);
  int v = r0 % Vv;
  int r = (r0 / Vv) % Rr;
  int n = (r0 / (Vv * Rr)) % Nn;
  int i = r0 / (Vv * Rr * Nn);
  const float* w = which ? w2 : w1;
  const float* b = which ? b2 : b1;
  const float* xm = ws + OFF_XM + (size_t)n * Cc * Vv + v;
  const float* wr = w + (size_t)(i * Rr + r) * Cc;
  float s = b[i * Rr + r];
  for (int c = 0; c < Cc; ++c) s += wr[c] * xm[c * Vv];
  ws[(which ? OFF_X2 : OFF_X1) + ((i * Nn + n) * Rr + r) * Vv + v] = s;
}

// adj[i,n,o,u,v] = alpha*(sum_r w4[i,o,r]*tanh(x1[i,n,r,u]-x2[i,n,r,v]) + b4[i,o]) + PA[i,u,v]
__global__ __launch_bounds__(256) void k3_adj(const float* __restrict__ w4, const float* __restrict__ b4,
                                              const float* __restrict__ PA, const float* __restrict__ alpha,
                                              float* __restrict__ ws) {
  __shared__ float s1[Rr * Vv], s2[Rr * Vv], corr[Rr * Vv * Vv];
  int i = blockIdx.x / Nn, n = blockIdx.x % Nn;
  int tid = threadIdx.x;
  if (tid < Rr * Vv) {
    s1[tid] = ws[OFF_X1 + (i * Nn + n) * Rr * Vv + tid];
    s2[tid] = ws[OFF_X2 + (i * Nn + n) * Rr * Vv + tid];
  }
  __syncthreads();
  for (int e = tid; e < Rr * Vv * Vv; e += 256) {
    int r = e / (Vv * Vv);
    int u = (e / Vv) % Vv;
    int v = e % Vv;
    corr[e] = tanhf(s1[r * Vv + u] - s2[r * Vv + v]);
  }
  __syncthreads();
  float al = alpha[0];
  for (int e = tid; e < Cc * Vv * Vv; e += 256) {
    int o = e / (Vv * Vv);
    int uv = e % (Vv * Vv);
    float s = b4[i * Cc + o];
    const float* wo = w4 + (i * Cc + o) * Rr;
#pragma unroll
    for (int r = 0; r < Rr; ++r) s += wo[r] * corr[r * Vv * Vv + uv];
    ws[OFF_ADJ + ((size_t)((i * Nn + n) * Cc + o)) * (Vv * Vv) + uv] = al * s + PA[i * Vv * Vv + uv];
  }
}

// Fused: per (n, 16-t tile): X3 = W3_i @ X (WMMA), then y[n,o,t,u] += X3_o @ adj_o^T (WMMA), summed over i.
__global__ __launch_bounds__(256) void k4_fused(const float* __restrict__ x,
                                                const float* __restrict__ w3, const float* __restrict__ b3,
                                                const float* __restrict__ adj, float* __restrict__ y) {
  __shared__ float sX3[64 * 400];                 // 100 KB
  __shared__ float sW[64 * 64];                   // 16 KB (TDM-staged W3_i)
  __shared__ float sA[8][640];                    // 20 KB per-wave adjacency staging
  int n = blockIdx.x >> 4;
  int t0 = (blockIdx.x & 15) * 16;
  int tid = threadIdx.x;
  int w = tid >> 5, lane = tid & 31;
  int l16 = lane & 15;
  int hi = lane >> 4;                             // 0|1
  int khalf = hi * 2;                             // A/B frag: lanes 16-31 carry K+2,K+3
  const float* xn = x + (size_t)n * CTV_;

  v8f acc[8][2];
#pragma unroll
  for (int a = 0; a < 8; ++a)
#pragma unroll
    for (int b = 0; b < 2; ++b)
#pragma unroll
      for (int r = 0; r < 8; ++r) acc[a][b][r] = 0.f;

  for (int i = 0; i < Ss; ++i) {
    // ---- TDM: stage W3_i (64x64) into LDS ----
    if (tid < 32) {
      tdm_load_1d(lds_off(sW), (unsigned long long)(w3 + i * 4096), 4096);
      __builtin_amdgcn_s_wait_tensorcnt((short)0);
    }
    __syncthreads();

    // ---- GEMM1: X3(64x400) = W3_i(64x64) @ Xtile(64x400) ----
    int mt = w >> 1;                              // m-tile 0..3, two waves each
    int m = mt * 16 + l16;
    for (int nt = (w & 1); nt < 25; nt += 2) {    // wave-uniform trip count
      int gcol = t0 * 25 + nt * 16 + l16;         // column in 64x6400 x[n]
      v2f bfr[16];
#pragma unroll
      for (int kq = 0; kq < 16; ++kq) {           // preload full K as one clause
        bfr[kq].x = xn[(size_t)(kq * 4 + khalf) * TV_ + gcol];
        bfr[kq].y = xn[(size_t)(kq * 4 + khalf + 1) * TV_ + gcol];
      }
      v8f c;
#pragma unroll
      for (int r = 0; r < 8; ++r) c[r] = 0.f;
#pragma unroll
      for (int kq = 0; kq < 16; ++kq) {
        v2f a;
        a.x = sW[m * 64 + kq * 4 + khalf];
        a.y = sW[m * 64 + kq * 4 + khalf + 1];
        c = __builtin_amdgcn_wmma_f32_16x16x4_f32(false, a, false, bfr[kq], (short)0, c, false, false);
      }
#pragma unroll
      for (int r = 0; r < 8; ++r) {               // C/D: vgpr r -> M = r (lanes0-15) / r+8 (lanes16-31)
        int mg = mt * 16 + r + hi * 8;
        sX3[mg * 400 + nt * 16 + l16] = c[r] + b3[i * 64 + mg];
      }
    }
    __syncthreads();

    // ---- Contraction: per o, Y(16x25) += X3_o(16x25) @ adjT(25x25); K padded to 28, N to 32 ----
#pragma unroll
    for (int oo = 0; oo < 8; ++oo) {
      int o = w * 8 + oo;
      const float* ao = adj + ((size_t)((i * 64 + n) * 64 + o)) * 625;
      float* aw = sA[w];
      for (int e = lane; e < 625; e += 32) aw[e] = ao[e];   // coalesced stage, wave-private
      const float* x3o = sX3 + o * 400;
#pragma unroll
      for (int ut = 0; ut < 2; ++ut) {
        v8f cc = acc[oo][ut];
        int u = ut * 16 + l16;
        bool uok = (u < 25);
#pragma unroll
        for (int kb = 0; kb < 25; kb += 4) {      // 7 steps cover K=0..27
          int v0 = kb + khalf;
          v2f a, b;
          a.x = (v0 < 25) ? x3o[l16 * 25 + v0] : 0.f;
          a.y = (v0 + 1 < 25) ? x3o[l16 * 25 + v0 + 1] : 0.f;
          b.x = (uok && v0 < 25) ? aw[u * 25 + v0] : 0.f;
          b.y = (uok && v0 + 1 < 25) ? aw[u * 25 + v0 + 1] : 0.f;
          cc = __builtin_amdgcn_wmma_f32_16x16x4_f32(false, a, false, b, (short)0, cc, false, false);
        }
        acc[oo][ut] = cc;
      }
    }
    __syncthreads();                              // before next i overwrites sX3/sW
  }

  // ---- store y tile ----
#pragma unroll
  for (int oo = 0; oo < 8; ++oo) {
    int o = w * 8 + oo;
#pragma unroll
    for (int ut = 0; ut < 2; ++ut) {
      int u = ut * 16 + l16;
      if (u < 25) {
#pragma unroll
        for (int r = 0; r < 8; ++r) {
          int tl = r + hi * 8;
          y[((size_t)(n * 64 + o) * 256 + t0 + tl) * 25 + u] = acc[oo][ut][r];
        }
      }
    }
  }
}

// BN1 stats: per-channel sum / sumsq over (N,T,V)
__global__ __launch_bounds__(256) void k5_bn1_stats(const float* __restrict__ y, float* __restrict__ st) {
  __shared__ float rs[256], rq[256];
  int c = blockIdx.x >> 5;
  int n0 = (blockIdx.x & 31) * 2;
  float s = 0.f, q = 0.f;
  for (int e = threadIdx.x; e < 2 * TV_; e += 256) {
    int n = n0 + e / TV_;
    float v = y[(size_t)(n * Cc + c) * TV_ + (e % TV_)];
    s += v; q += v * v;
  }
  rs[threadIdx.x] = s; rq[threadIdx.x] = q;
  __syncthreads();
  for (int w = 128; w > 0; w >>= 1) {
    if (threadIdx.x < w) { rs[threadIdx.x] += rs[threadIdx.x + w]; rq[threadIdx.x] += rq[threadIdx.x + w]; }
    __syncthreads();
  }
  if (threadIdx.x == 0) { atomicAdd(&st[c], rs[0]); atomicAdd(&st[64 + c], rq[0]); }
}

// yr = relu(bn1(y)+x); accumulate GroupNorm stats per (n,g)
__global__ __launch_bounds__(256) void k6_bn1_apply(const float* __restrict__ x,
                                                    const float* __restrict__ bn_g, const float* __restrict__ bn_b,
                                                    const float* __restrict__ y, float* __restrict__ yr,
                                                    float* __restrict__ st) {
  __shared__ float rs[256], rq[256];
  int n = blockIdx.x >> 3, g = blockIdx.x & 7;
  const float inv = 1.f / (Nn * TV_);
  float s = 0.f, q = 0.f;
  for (int e = threadIdx.x; e < CGc * TV_; e += 256) {
    int c = g * CGc + e / TV_;
    size_t idx = (size_t)(n * Cc + c) * TV_ + (e % TV_);
    float mu = st[c] * inv;
    float var = st[64 + c] * inv - mu * mu;
    float val = (y[idx] - mu) * rsqrtf(var + 1e-5f) * bn_g[c] + bn_b[c] + x[idx];
    val = fmaxf(val, 0.f);
    yr[idx] = val;
    s += val; q += val * val;
  }
  rs[threadIdx.x] = s; rq[threadIdx.x] = q;
  __syncthreads();
  for (int w = 128; w > 0; w >>= 1) {
    if (threadIdx.x < w) { rs[threadIdx.x] += rs[threadIdx.x + w]; rq[threadIdx.x] += rq[threadIdx.x + w]; }
    __syncthreads();
  }
  if (threadIdx.x == 0) { atomicAdd(&st[128 + n * 8 + g], rs[0]); atomicAdd(&st[640 + n * 8 + g], rq[0]); }
}

// GroupNorm apply + grouped (9,1) temporal conv + BN2 stats; yr window staged via TDM
__global__ __launch_bounds__(256) void k7_gn_conv(const float* __restrict__ yr,
                                                  const float* __restrict__ gn_g, const float* __restrict__ gn_b,
                                                  const float* __restrict__ tw, const float* __restrict__ tb,
                                                  float* __restrict__ z, float* __restrict__ st) {
  __shared__ float sT[CGc * 72 * Vv];             // 8ch x (64+8)t x 25v = 57.6 KB
  __shared__ float sSum[CGc], sSq[CGc];
  int b = blockIdx.x;
  int n = b / (Gg * 4);
  int g = (b / 4) % Gg;
  int t0 = (b % 4) * 64;
  int tid = threadIdx.x;
  if (tid < CGc) { sSum[tid] = 0.f; sSq[tid] = 0.f; }
  // zero halo (and everything) so conv padding stays zero after in-place GN
  for (int e = tid; e < CGc * 72 * Vv; e += 256) sT[e] = 0.f;
  __syncthreads();

  int tlo = t0 - 4; if (tlo < 0) tlo = 0;
  int thi = t0 + 68; if (thi > Tt) thi = Tt;
  int len = (thi - tlo) * Vv;                     // <= 1800 elements per channel row
  int shift = tlo - (t0 - 4);                     // 0 or 4 t-slots of leading pad
  if (tid < 32) {                                 // wave 0 issues 8 row-tiles via TDM
    unsigned base = lds_off(sT);
    for (int cl = 0; cl < CGc; ++cl) {
      unsigned long long ga =
          (unsigned long long)(yr + (size_t)(n * Cc + g * CGc + cl) * TV_ + (size_t)tlo * Vv);
      tdm_load_1d(base + (unsigned)((cl * 72 * Vv + shift * Vv) * 4), ga, len);
    }
    __builtin_amdgcn_s_wait_tensorcnt((short)0);
  }
  __syncthreads();

  const float inv = 1.f / (CGc * TV_);
  float mu = st[128 + n * 8 + g] * inv;
  float var = st[640 + n * 8 + g] * inv - mu * mu;
  float rsg = rsqrtf(var + 1e-5f);
  // in-place GroupNorm on the valid interior only (halo stays zero)
  for (int e = tid; e < CGc * 72 * Vv; e += 256) {
    int cl = e / (72 * Vv);
    int rr = e % (72 * Vv);
    int tt = rr / Vv;
    int tin = t0 - 4 + tt;
    if (tin >= 0 && tin < Tt) {
      int c = g * CGc + cl;
      sT[e] = (sT[e] - mu) * rsg * gn_g[c] + gn_b[c];
    }
  }
  __syncthreads();

  for (int e = tid; e < CGc * 64 * Vv; e += 256) {
    int ol = e / (64 * Vv);
    int rr = e % (64 * Vv);
    int tt = rr / Vv, v = rr % Vv;
    int o = g * CGc + ol;
    float acc = tb[o];
    const float* two = tw + o * CGc * KK;
#pragma unroll
    for (int cl = 0; cl < CGc; ++cl) {
      const float* base = &sT[cl * 72 * Vv + tt * Vv + v];
#pragma unroll
      for (int k = 0; k < KK; ++k) acc += two[cl * KK + k] * base[k * Vv];
    }
    z[(size_t)(n * Cc + o) * TV_ + (t0 + tt) * Vv + v] = acc;
    atomicAdd(&sSum[ol], acc);
    atomicAdd(&sSq[ol], acc * acc);
  }
  __syncthreads();
  if (tid < CGc) {
    atomicAdd(&st[1152 + g * CGc + tid], sSum[tid]);
    atomicAdd(&st[1216 + g * CGc + tid], sSq[tid]);
  }
}

// out = relu(bn2(z) + x)
__global__ __launch_bounds__(256) void k8_final(const float* __restrict__ x,
                                                const float* __restrict__ bn2_g, const float* __restrict__ bn2_b,
                                                const float* __restrict__ z, const float* __restrict__ st,
                                                float* __restrict__ out) {
  int idx = blockIdx.x * 256 + threadIdx.x;
  if (idx >= Nn * Cc * TV_) return;
  int c = (idx / TV_) % Cc;
  const float inv = 1.f / (Nn * TV_);
  float mu = st[1152 + c] * inv;
  float var = st[1216 + c] * inv - mu * mu;
  float val = (z[idx] - mu) * rsqrtf(var + 1e-5f) * bn2_g[c] + bn2_b[c] + x[idx];
  out[idx] = fmaxf(val, 0.f);
}

extern "C" void kernel_launch(void* const* d_in, const int* in_sizes, int n_in,
                              void* d_out, int out_size, void* d_ws, size_t ws_size,
                              hipStream_t stream) {
  const float* x     = (const float*)d_in[0];
  const float* w1    = (const float*)d_in[1];
  const float* b1    = (const float*)d_in[2];
  const float* w2    = (const float*)d_in[3];
  const float* b2    = (const float*)d_in[4];
  const float* w3    = (const float*)d_in[5];
  const float* b3    = (const float*)d_in[6];
  const float* w4    = (const float*)d_in[7];
  const float* b4    = (const float*)d_in[8];
  const float* PA    = (const float*)d_in[9];
  const float* alpha = (const float*)d_in[10];
  const float* bn_g  = (const float*)d_in[11];
  const float* bn_b  = (const float*)d_in[12];
  const float* gn_g  = (const float*)d_in[13];
  const float* gn_b  = (const float*)d_in[14];
  const float* tw    = (const float*)d_in[15];
  const float* tb    = (const float*)d_in[16];
  const float* bn2_g = (const float*)d_in[17];
  const float* bn2_b = (const float*)d_in[18];
  float* ws = (float*)d_ws;
  float* st = ws + OFF_ST;
  float* out = (float*)d_out;

  hipLaunchKernelGGL(k0_zero,      dim3(5),           dim3(256), 0, stream, st);
  hipLaunchKernelGGL(k1_mean,      dim3(400),         dim3(256), 0, stream, x, ws);
  hipLaunchKernelGGL(k2_x12,       dim3(300),         dim3(256), 0, stream, w1, b1, w2, b2, ws);
  hipLaunchKernelGGL(k3_adj,       dim3(Ss * Nn),     dim3(256), 0, stream, w4, b4, PA, alpha, ws);
  hipLaunchKernelGGL(k4_fused,     dim3(Nn * 16),     dim3(256), 0, stream, x, w3, b3, ws + OFF_ADJ, ws + OFF_Y);
  hipLaunchKernelGGL(k5_bn1_stats, dim3(Cc * 32),     dim3(256), 0, stream, ws + OFF_Y, st);
  hipLaunchKernelGGL(k6_bn1_apply, dim3(Nn * Gg),     dim3(256), 0, stream, x, bn_g, bn_b, ws + OFF_Y, ws + OFF_YR, st);
  hipLaunchKernelGGL(k7_gn_conv,   dim3(Nn * Gg * 4), dim3(256), 0, stream, ws + OFF_YR, gn_g, gn_b, tw, tb, ws + OFF_ADJ, st);
  hipLaunchKernelGGL(k8_final,     dim3(25600),       dim3(256), 0, stream, x, bn2_g, bn2_b, ws + OFF_ADJ, st, out);
}